// Attention_81836306858184
// MI455X (gfx1250) — compile-verified
//
#include <hip/hip_runtime.h>
#include <math.h>

#define B_  8
#define I_  16
#define C_  256
#define HW_ 4096
#define NP  4             // pixel tiles resident per block in k_main
#define PR  32            // dwords per channel-pair row (16 px * 2)
#define TSZ 4096          // floats per tile buffer (128 pair-rows * 32)
#define CP  17            // padded stride for epilogue staging only

typedef float v2f __attribute__((ext_vector_type(2)));
typedef float v8f __attribute__((ext_vector_type(8)));

// ---------------------------------------------------------------------------
// K0: transpose 256x256 weights so weight-column (breg) loads are coalesced.
// ---------------------------------------------------------------------------
__global__ void k_transpose(const float* __restrict__ Wf, const float* __restrict__ Wo,
                            float* __restrict__ WfT, float* __restrict__ WoT) {
    int idx = blockIdx.x * blockDim.x + threadIdx.x;   // 0..65535
    int c = idx >> 8, o = idx & 255;
    WfT[idx] = Wf[o * C_ + c];
    WoT[idx] = Wo[o * C_ + c];
}

// ---------------------------------------------------------------------------
// K1: per (b,i) softmax stats. Koff[b,i] = M + log(Z),
//   M = max_{c,hw}(m*Wm+bm) via sign-aware min/max of masks,
//   Z = sum_{c,hw} exp(m*Wm+bm-M).
// ---------------------------------------------------------------------------
__global__ void __launch_bounds__(256)
k_stats(const float* __restrict__ masks, const float* __restrict__ Wm,
        const float* __restrict__ bm, float* __restrict__ Koff) {
    __shared__ float red[256];
    __shared__ float wm_s[256];
    __shared__ float bmM_s[256];

    const int b = blockIdx.x / I_, i = blockIdx.x % I_;
    const int tid = threadIdx.x;
    const float* mp = masks + (size_t)(b * I_ + i) * HW_;

    float mloc[16];
    float mx = -1e30f, mn = 1e30f;
#pragma unroll
    for (int j = 0; j < 16; ++j) {
        float v = mp[tid + j * 256];
        mloc[j] = v;
        mx = fmaxf(mx, v);
        mn = fminf(mn, v);
    }
    red[tid] = mx; __syncthreads();
    for (int s = 128; s > 0; s >>= 1) { if (tid < s) red[tid] = fmaxf(red[tid], red[tid + s]); __syncthreads(); }
    mx = red[0]; __syncthreads();
    red[tid] = mn; __syncthreads();
    for (int s = 128; s > 0; s >>= 1) { if (tid < s) red[tid] = fminf(red[tid], red[tid + s]); __syncthreads(); }
    mn = red[0]; __syncthreads();

    const float w  = Wm[i * C_ + tid];
    const float bb = bm[i * C_ + tid];
    wm_s[tid] = w;
    red[tid] = bb + fmaxf(w * mx, w * mn); __syncthreads();
    for (int s = 128; s > 0; s >>= 1) { if (tid < s) red[tid] = fmaxf(red[tid], red[tid + s]); __syncthreads(); }
    const float M = red[0]; __syncthreads();
    bmM_s[tid] = bb - M;
    __syncthreads();

    float z = 0.f;
    for (int j = 0; j < 16; ++j) {
        float m = mloc[j];
#pragma unroll 8
        for (int c = 0; c < 256; ++c)
            z += __expf(fmaf(m, wm_s[c], bmM_s[c]));
    }
    red[tid] = z; __syncthreads();
    for (int s = 128; s > 0; s >>= 1) { if (tid < s) red[tid] += red[tid + s]; __syncthreads(); }
    if (tid == 0) Koff[b * I_ + i] = M + __logf(red[0]);
}

// ---------------------------------------------------------------------------
// K2: fused main pass, weight-stationary.
//   8 waves/block; wave owns o-tiles {2*wv, 2*wv+1}; weight column (64 k-steps)
//   in 128 VGPRs, reused over NP=4 LDS-resident pixel tiles.
//   Tiles stored paired-channel interleaved: (c,p) -> (c>>1)*32 + 2p + (c&1),
//   so each WMMA A-fragment is ONE aligned ds_load_b64, conflict-free.
// ---------------------------------------------------------------------------
__global__ void __launch_bounds__(256, 4)
k_main(const float* __restrict__ x, const float* __restrict__ masks,
       const float* __restrict__ WfT, const float* __restrict__ bf,
       const float* __restrict__ Wm, const float* __restrict__ bm,
       const float* __restrict__ WoT, const float* __restrict__ bo,
       const float* __restrict__ gptr, const float* __restrict__ Koff,
       float* __restrict__ out) {
    // 135,232 bytes static LDS (compile-verified headroom on gfx1250)
    __shared__ float smem[NP * TSZ * 2 + NP * 256 + 16];
    float* xs = smem;                       // NP x-tiles (later: epilogue staging)
    float* fs = smem + NP * TSZ;            // NP feat/g tiles
    float* mt = smem + 2 * NP * TSZ;        // NP x [i*16+p] mask tiles
    float* sK = mt + NP * 256;              // Koff[b, 0..15]

    const int tid  = threadIdx.x;
    const int wv   = tid >> 5;              // 0..7
    const int lane = tid & 31;
    const int pl   = lane & 15;
    const int half = lane >> 4;

    const int blocks_per_b = HW_ / (NP * 16);            // 64
    const int b     = blockIdx.x / blocks_per_b;
    const int blkp0 = (blockIdx.x % blocks_per_b) * (NP * 16);

    const float gamma = gptr[0];
    if (tid < I_) sK[tid] = Koff[b * I_ + tid];

    const float* xb = x + (size_t)b * C_ * HW_;
    // ---- stage x tiles: float4 global -> paired-channel LDS ----
#pragma unroll
    for (int j = 0; j < 16; ++j) {
        int lin = (tid + 256 * j) * 4;                    // 0..16380
        int t = lin >> 12, rem = lin & 4095;
        int c = rem >> 4, p = rem & 15;
        const float4 v = *(const float4*)(xb + (size_t)c * HW_ + blkp0 + t * 16 + p);
        float* d = xs + t * TSZ + (c >> 1) * PR + (c & 1);
        d[2 * (p + 0)] = v.x; d[2 * (p + 1)] = v.y;
        d[2 * (p + 2)] = v.z; d[2 * (p + 3)] = v.w;
    }
    // ---- stage mask tiles ----
    {
        const float* mb = masks + (size_t)b * I_ * HW_;
#pragma unroll
        for (int j = 0; j < NP; ++j) {
            int idx = tid + 256 * j;                      // 0..1023
            int t = idx >> 8, rem = idx & 255;
            mt[idx] = mb[(rem >> 4) * HW_ + blkp0 + t * 16 + (rem & 15)];
        }
    }
    __syncthreads();

    v2f breg[64];

    // ================= matmul1: feat = x @ Wf^T + bf =================
    for (int oj = 0; oj < 2; ++oj) {
        const int o0 = (wv * 2 + oj) * 16;
        __builtin_prefetch(WoT + o0 + pl, 0, 0);          // warm next-phase column
#pragma unroll
        for (int k = 0; k < 64; ++k) {
            int c = 4 * k + 2 * half;
            breg[k].x = WfT[c * C_ + o0 + pl];
            breg[k].y = WfT[(c + 1) * C_ + o0 + pl];
        }
        const float bias = bf[o0 + pl];
        for (int t = 0; t < NP; ++t) {
            const float* a = xs + t * TSZ + half * PR + 2 * pl;  // A base (8B aligned)
            v8f acc = {0.f, 0.f, 0.f, 0.f, 0.f, 0.f, 0.f, 0.f};
#pragma unroll
            for (int k = 0; k < 64; ++k) {
                const v2f af = *(const v2f*)(a + 64 * k);        // one ds_load_b64
                acc = __builtin_amdgcn_wmma_f32_16x16x4_f32(false, af, false, breg[k],
                                                            (short)0, acc, false, false);
            }
            float* f = fs + t * TSZ + o0 * 16 + (pl >> 1) * PR + (pl & 1) + 16 * half;
#pragma unroll
            for (int r = 0; r < 8; ++r)
                f[2 * r] = acc[r] + bias;                 // (o0+pl, r+8*half)
        }
    }
    __syncthreads();

    // ========= msum: g = feat * sum_i exp(m*Wm + bm - Koff), p-major lanes ====
    {
        const int p  = tid & 15;
        const int cg = tid >> 4;                          // 0..15
        for (int j = 0; j < 16; ++j) {
            const int c = cg * 16 + j;
            float warr[16], aarr[16];
#pragma unroll
            for (int i = 0; i < 16; ++i) {
                warr[i] = Wm[i * C_ + c];                 // half-wave broadcast
                aarr[i] = bm[i * C_ + c] - sK[i];
            }
            const int gaddr = (c >> 1) * PR + 2 * p + (c & 1);   // 2-way max
            for (int t = 0; t < NP; ++t) {
                float acc = 0.f;
#pragma unroll
                for (int i = 0; i < 16; ++i)
                    acc += __expf(fmaf(mt[t * 256 + i * 16 + p], warr[i], aarr[i]));
                fs[t * TSZ + gaddr] *= acc;
            }
        }
    }
    __syncthreads();

    // ================= matmul2 + epilogue: gamma*(g@Wo^T + 16*bo) + x =======
    for (int oj = 0; oj < 2; ++oj) {
        const int o0 = (wv * 2 + oj) * 16;
#pragma unroll
        for (int k = 0; k < 64; ++k) {
            int c = 4 * k + 2 * half;
            breg[k].x = WoT[c * C_ + o0 + pl];
            breg[k].y = WoT[(c + 1) * C_ + o0 + pl];
        }
        const float bias = 16.0f * bo[o0 + pl];
        for (int t = 0; t < NP; ++t) {
            const float* a = fs + t * TSZ + half * PR + 2 * pl;
            v8f acc = {0.f, 0.f, 0.f, 0.f, 0.f, 0.f, 0.f, 0.f};
#pragma unroll
            for (int k = 0; k < 64; ++k) {
                const v2f af = *(const v2f*)(a + 64 * k);        // one ds_load_b64
                acc = __builtin_amdgcn_wmma_f32_16x16x4_f32(false, af, false, breg[k],
                                                            (short)0, acc, false, false);
            }
            // x tiles are dead: reuse xs[t] as per-wave transpose staging
            float* ost = xs + t * TSZ + wv * 272;         // 16*17 floats, wave-private
#pragma unroll
            for (int r = 0; r < 8; ++r)
                ost[pl * CP + r + 8 * half] = acc[r] + bias;
            // in-wave DS ordering: no barrier needed
            const int p0t = blkp0 + t * 16;
#pragma unroll
            for (int h = 0; h < 2; ++h) {
                const int oloc = 8 * h + (lane >> 2);
                const int p4   = (lane & 3) * 4;
                float r0 = ost[oloc * CP + p4 + 0];
                float r1 = ost[oloc * CP + p4 + 1];
                float r2 = ost[oloc * CP + p4 + 2];
                float r3 = ost[oloc * CP + p4 + 3];
                const size_t off = (size_t)(o0 + oloc) * HW_ + p0t + p4;
                const float4 x4 = *(const float4*)(xb + off);
                float4 res;
                res.x = fmaf(gamma, r0, x4.x);
                res.y = fmaf(gamma, r1, x4.y);
                res.z = fmaf(gamma, r2, x4.z);
                res.w = fmaf(gamma, r3, x4.w);
                *(float4*)(out + (size_t)b * C_ * HW_ + off) = res;  // coalesced
            }
        }
    }
}

// ---------------------------------------------------------------------------
extern "C" void kernel_launch(void* const* d_in, const int* in_sizes, int n_in,
                              void* d_out, int out_size, void* d_ws, size_t ws_size,
                              hipStream_t stream) {
    const float* x     = (const float*)d_in[0];
    const float* masks = (const float*)d_in[1];
    const float* Wf    = (const float*)d_in[2];
    const float* bf    = (const float*)d_in[3];
    const float* Wm    = (const float*)d_in[4];
    const float* bm    = (const float*)d_in[5];
    const float* Wo    = (const float*)d_in[6];
    const float* bo    = (const float*)d_in[7];
    const float* gamma = (const float*)d_in[8];

    float* ws   = (float*)d_ws;
    float* WfT  = ws;                 // 65536 floats
    float* WoT  = ws + 65536;         // 65536 floats
    float* Koff = ws + 131072;        // 128 floats

    k_transpose<<<256, 256, 0, stream>>>(Wf, Wo, WfT, WoT);
    k_stats<<<B_ * I_, 256, 0, stream>>>(masks, Wm, bm, Koff);
    k_main<<<(B_ * HW_) / (NP * 16), 256, 0, stream>>>(
        x, masks, WfT, bf, Wm, bm, WoT, bo, gamma, Koff, (float*)d_out);
}